// SwitchFocusedLoss3Class_80221399155357
// MI455X (gfx1250) — compile-verified
//
#include <hip/hip_runtime.h>
#include <stdint.h>

// SwitchFocusedLoss3Class for MI455X (gfx1250).
//
// Roofline: 160 MiB read / 4 B written -> HBM-bandwidth problem (~7us at
// 23.3 TB/s). No matrix contraction exists, so WMMA is not applicable; the
// CDNA5 features that matter are the async global->LDS copy path (ASYNCcnt)
// for staging the labels row, global_prefetch_b8 for the streaming logits,
// and wave32-sized blocks. Per-element VALU work is minimized so the kernel
// sits on the memory roofline:
//   - proximity window via per-thread is_switch bitmask + two 32-entry LDS
//     product LUTs (bank-conflict-free)
//   - 2 hardware exps (v_exp_f32 via __expf) + 1 v_log_f32 (__logf);
//     exp(max-max)==1 exactly so only two exps are needed
//   - float4/int4 chunked streams (each thread owns 32 contiguous positions)

#define S_LEN   8192
#define THREADS 256   // 8 wave32 waves per block; 32 positions per thread
#define CHUNK   (S_LEN / THREADS)   // 32

__device__ __forceinline__ void accum_pos(
    float l0, float l1, float l2, int lb, int sg, int j,
    uint64_t Rbits, uint64_t Lbits,
    const float* __restrict__ lutR, const float* __restrict__ lutL,
    float& acc, int& cnt)
{
    const bool valid = (lb != -100);

    // argmax over C=3, first maximum wins (jnp.argmax semantics)
    int pred = 0; float best = l0;
    if (l1 > best) { best = l1; pred = 1; }
    if (l2 > best) { best = l2; pred = 2; }

    // log-sum-exp: exp(l_max - m) == 1 exactly, only 2 hardware exps
    const float a  = (pred == 0) ? l1 : l0;
    const float b  = (pred == 2) ? l1 : l2;
    const float se = 1.0f + __expf(a - best) + __expf(b - best);
    const float lse = best + __logf(se);

    const int   slb  = valid ? lb : 0;
    const float lsel = (slb == 0) ? l0 : ((slb == 1) ? l1 : l2);
    const float ce   = valid ? (lse - lsel) : 0.0f;

    const bool pred_sw = (pred != 0);
    const bool is_swb  = ((unsigned)(lb - 1)) <= 1u;   // lb==1 || lb==2

    // proximity products from bitmask windows (LUT = product of factors)
    const float pr = lutR[(uint32_t)(Rbits >> (j + 1)) & 31u];
    const float pl = lutL[(uint32_t)(Lbits >> (27 + j)) & 31u];

    // w = (mult?pr:1) * (is_swb ? base : (mult?pl:1)), then seg factor.
    // Equivalent to the reference: at true-switch positions the offset-0
    // branch ASSIGNS base (left factors dropped), only right ones multiply;
    // elsewhere both sides multiply; factors gated on pred in {1,2} & valid.
    const bool  mult = pred_sw && valid;
    const float t1   = mult ? pr : 1.0f;
    const float base = pred_sw ? 10.0f : 20.0f;        // SWITCH_RECALL_WEIGHT
    const float t2   = (is_swb && valid) ? base : (mult ? pl : 1.0f);
    float w = t1 * t2;

    if ((sg > 0) && valid && !is_swb)                  // seg adjustment
        w *= pred_sw ? 3.0f : 0.3f;                    // SEG_PENALTY / SEG_REWARD

    acc = fmaf(ce, w, acc);
    cnt += valid ? 1 : 0;
}

__global__ __launch_bounds__(THREADS)
void switch_loss_rows(const float* __restrict__ logits,
                      const int*   __restrict__ labels,
                      const int*   __restrict__ seg,
                      float* __restrict__ part_sum,
                      int*   __restrict__ part_cnt)
{
    __shared__ __align__(16) int lab[S_LEN];   // 32 KB: whole label row
    __shared__ uint32_t smask[THREADS];        // per-thread is_switch bitmasks
    __shared__ float    lutR[32];              // product LUT, distance k+1 per bit k
    __shared__ float    lutL[32];              // product LUT, distance 5-k per bit k
    __shared__ float    red_f[THREADS];
    __shared__ int      red_i[THREADS];

    const int t = threadIdx.x;
    const int b = blockIdx.x;
    const long rowBase = (long)b * S_LEN;
    const int*   labRow = labels + rowBase;
    const int*   segRow = seg    + rowBase;
    const float* logRow = logits + rowBase * 3;

    // ---- Stage labels row into LDS via CDNA5 async global->LDS B128 copies.
    // Low 32 bits of a flat shared-pointer == LDS byte offset (ISA aperture
    // rule), which is what VDST of the async op wants.
    {
        const uint32_t ldsBase = (uint32_t)(uintptr_t)(&lab[0]);
        #pragma unroll
        for (int k = 0; k < S_LEN / 4 / THREADS; ++k) {
            const uint32_t voff  = (uint32_t)(t + THREADS * k) * 16u;
            const uint32_t laddr = ldsBase + voff;
            asm volatile("global_load_async_to_lds_b128 %0, %1, %2"
                         :
                         : "v"(laddr), "v"(voff), "s"(labRow)
                         : "memory");
        }
    }

    // ---- Proximity-factor LUTs (independent of staged data).
    // f[d] = 2 - d/(PROX_TOL+1); exp(sum log f) == prod f exactly.
    if (t < 32) {
        const float f1 = 11.0f / 6.0f, f2 = 10.0f / 6.0f, f3 = 1.5f,
                    f4 = 4.0f / 3.0f,  f5 = 7.0f / 6.0f;
        float pr = 1.0f, pl = 1.0f;
        if (t & 1)  { pr *= f1; pl *= f5; }
        if (t & 2)  { pr *= f2; pl *= f4; }
        if (t & 4)  { pr *= f3; pl *= f3; }
        if (t & 8)  { pr *= f4; pl *= f2; }
        if (t & 16) { pr *= f5; pl *= f1; }
        lutR[t] = pr;
        lutL[t] = pl;
    }

    asm volatile("s_wait_asynccnt 0" ::: "memory");
    __syncthreads();                       // labels + LUTs visible to all waves

    // ---- Build per-thread is_switch bitmask for the owned 32-position chunk.
    const int c0 = t * CHUNK;
    {
        const int4* labV = (const int4*)(&lab[c0]);
        uint32_t m = 0;
        #pragma unroll
        for (int g = 0; g < CHUNK / 4; ++g) {
            const int4 v = labV[g];
            m |= (((unsigned)(v.x - 1) <= 1u) ? 1u : 0u) << (4 * g + 0);
            m |= (((unsigned)(v.y - 1) <= 1u) ? 1u : 0u) << (4 * g + 1);
            m |= (((unsigned)(v.z - 1) <= 1u) ? 1u : 0u) << (4 * g + 2);
            m |= (((unsigned)(v.w - 1) <= 1u) ? 1u : 0u) << (4 * g + 3);
        }
        smask[t] = m;
    }
    __syncthreads();

    // ---- Main loop: 32 contiguous positions per thread, vectorized streams.
    const uint32_t w_prev = (t > 0)           ? smask[t - 1] : 0u;
    const uint32_t w_cur  = smask[t];
    const uint32_t w_next = (t < THREADS - 1) ? smask[t + 1] : 0u;
    // Rbits bit m = is_switch at chunk position m    (read bits j+1 .. j+5)
    // Lbits bit m = is_switch at chunk position m-32 (read bits 27+j .. 31+j)
    const uint64_t Rbits = (uint64_t)w_cur  | ((uint64_t)w_next << 32);
    const uint64_t Lbits = (uint64_t)w_prev | ((uint64_t)w_cur  << 32);

    const float4* logV = (const float4*)(logRow + (size_t)c0 * 3);  // 384B chunk
    const int4*   segV = (const int4*)(segRow + c0);
    const int4*   labV = (const int4*)(&lab[c0]);

    float acc = 0.0f;
    int   cnt = 0;

    #pragma unroll 4
    for (int g = 0; g < CHUNK / 4; ++g) {
        // pull the logits stream ahead through GL2 (global_prefetch_b8);
        // speculative prefetch past the buffer end is silently dropped.
        __builtin_prefetch((const char*)logV + 48 * (g + 2), 0, 3);

        const float4 A = logV[3 * g + 0];
        const float4 B = logV[3 * g + 1];
        const float4 C = logV[3 * g + 2];
        const int4 lb4 = labV[g];
        const int4 sg4 = segV[g];
        const int  j0  = 4 * g;

        accum_pos(A.x, A.y, A.z, lb4.x, sg4.x, j0 + 0, Rbits, Lbits, lutR, lutL, acc, cnt);
        accum_pos(A.w, B.x, B.y, lb4.y, sg4.y, j0 + 1, Rbits, Lbits, lutR, lutL, acc, cnt);
        accum_pos(B.z, B.w, C.x, lb4.z, sg4.z, j0 + 2, Rbits, Lbits, lutR, lutL, acc, cnt);
        accum_pos(C.y, C.z, C.w, lb4.w, sg4.w, j0 + 3, Rbits, Lbits, lutR, lutL, acc, cnt);
    }

    // ---- deterministic fixed-order block reduction
    red_f[t] = acc;
    red_i[t] = cnt;
    __syncthreads();
    for (int off = THREADS / 2; off > 0; off >>= 1) {
        if (t < off) {
            red_f[t] += red_f[t + off];
            red_i[t] += red_i[t + off];
        }
        __syncthreads();
    }
    if (t == 0) {
        part_sum[b] = red_f[0];
        part_cnt[b] = red_i[0];
    }
}

__global__ __launch_bounds__(256)
void final_reduce(const float* __restrict__ part_sum,
                  const int*   __restrict__ part_cnt,
                  int nblocks,
                  float* __restrict__ out)
{
    __shared__ float rf[256];
    __shared__ int   ri[256];
    const int t = threadIdx.x;
    float a = 0.0f;
    int   c = 0;
    for (int i = t; i < nblocks; i += 256) {
        a += part_sum[i];
        c += part_cnt[i];
    }
    rf[t] = a;
    ri[t] = c;
    __syncthreads();
    for (int off = 128; off > 0; off >>= 1) {
        if (t < off) {
            rf[t] += rf[t + off];
            ri[t] += ri[t + off];
        }
        __syncthreads();
    }
    if (t == 0) {
        const int d = (ri[0] > 1) ? ri[0] : 1;
        out[0] = rf[0] / (float)d;
    }
}

extern "C" void kernel_launch(void* const* d_in, const int* in_sizes, int n_in,
                              void* d_out, int out_size, void* d_ws, size_t ws_size,
                              hipStream_t stream)
{
    const float* logits = (const float*)d_in[0];
    const int*   labels = (const int*)d_in[1];
    const int*   segm   = (const int*)d_in[2];

    const int BS = in_sizes[1];           // B * S
    const int B  = BS / S_LEN;            // S fixed at 8192 by the reference

    float* part_sum = (float*)d_ws;
    int*   part_cnt = (int*)((char*)d_ws + (size_t)B * sizeof(float));

    switch_loss_rows<<<B, THREADS, 0, stream>>>(logits, labels, segm,
                                                part_sum, part_cnt);
    final_reduce<<<1, 256, 0, stream>>>(part_sum, part_cnt, B, (float*)d_out);
}